// Attention_17575006175247
// MI455X (gfx1250) — compile-verified
//
#include <hip/hip_runtime.h>
#include <cstdint>
#include <cstddef>

// ---------------------------------------------------------------------------
// Problem constants (match reference)
// ---------------------------------------------------------------------------
#define B_      64
#define DIM_    2048
#define HQ_     16
#define HK_     4
#define GQ_     4          // HQ / HK
#define D_      128
#define SPAST_  4095
#define SFULL_  4096       // SPAST_ + 1 appended token; 256 tiles of 16 exactly
#define NSPLIT_ 8
#define TILES_PER_SPLIT_ 32   // (SFULL_/16)/NSPLIT_

typedef float v2f __attribute__((ext_vector_type(2)));
typedef float v8f __attribute__((ext_vector_type(8)));

__device__ __forceinline__ v8f wmma_f32(v2f a, v2f b, v8f c) {
  // V_WMMA_F32_16X16X4_F32 : D = A(16x4) x B(4x16) + C(16x16)
  return __builtin_amdgcn_wmma_f32_16x16x4_f32(false, a, false, b, (short)0, c,
                                               false, false);
}

// CDNA5 async global->LDS copy (ASYNCcnt-tracked), 16 bytes per lane.
// VDST VGPR holds the LDS byte address (low 32 bits of the generic pointer:
// flat->LDS aperture mapping truncates to the 32-bit LDS offset).
__device__ __forceinline__ void async_load_f4_to_lds(uint32_t lds_byte_addr,
                                                     const float* gaddr) {
  asm volatile("global_load_async_to_lds_b128 %0, %1, off"
               :
               : "v"(lds_byte_addr), "v"(gaddr)
               : "memory");
}
__device__ __forceinline__ void wait_asynccnt0() {
  asm volatile("s_wait_asynccnt 0x0" ::: "memory");
}

// ---------------------------------------------------------------------------
// GEMM: C[M,N] = A[M,K] * W[N,K]^T   (one wave per 16x16 tile)
// A fragment: lane<16 -> row l16, {K+0,K+1}; lane>=16 -> row l16, {K+2,K+3}
// B fragment: lane<16 -> col l16, {K+0,K+1}; lane>=16 -> col l16, {K+2,K+3}
// C fragment: vgpr r -> row (r + 8*half), col l16
// ---------------------------------------------------------------------------
__global__ void __launch_bounds__(32)
gemm16_tn(const float* __restrict__ A, const float* __restrict__ W,
          float* __restrict__ C, int M, int N, int K) {
  const int lane = threadIdx.x;
  const int l16  = lane & 15;
  const int half = lane >> 4;
  const int n0   = blockIdx.x * 16;
  const int m0   = blockIdx.y * 16;

  const float* arow = A + (size_t)(m0 + l16) * K + 2 * half;
  const float* wrow = W + (size_t)(n0 + l16) * K + 2 * half;

  v8f acc = {0.f, 0.f, 0.f, 0.f, 0.f, 0.f, 0.f, 0.f};
#pragma unroll 8
  for (int k0 = 0; k0 < K; k0 += 4) {
    v2f a; a.x = arow[k0]; a.y = arow[k0 + 1];
    v2f b; b.x = wrow[k0]; b.y = wrow[k0 + 1];
    acc = wmma_f32(a, b, acc);
  }
#pragma unroll
  for (int r = 0; r < 8; ++r)
    C[(size_t)(m0 + r + 8 * half) * N + n0 + l16] = acc[r];
}

// ---------------------------------------------------------------------------
// RMSNorm + RoPE on one 128-d head vector per wave.
// grid = (B, HQ+HK); y < HQ -> q head, else k head. In-place.
// ---------------------------------------------------------------------------
__global__ void __launch_bounds__(32)
normrope_kernel(float* __restrict__ qb, float* __restrict__ kb,
                const float* __restrict__ qw, const float* __restrict__ kw,
                const int* __restrict__ pos) {
  const int b    = blockIdx.x;
  const int hy   = blockIdx.y;
  const int lane = threadIdx.x;
  const bool isq = (hy < HQ_);

  float* vec = isq ? (qb + ((size_t)b * HQ_ + hy) * D_)
                   : (kb + ((size_t)b * HK_ + (hy - HQ_)) * D_);
  const float* w = isq ? qw : kw;

  float4 t = *(const float4*)(vec + lane * 4);
  float ss = t.x * t.x + t.y * t.y + t.z * t.z + t.w * t.w;
#pragma unroll
  for (int m = 1; m <= 16; m <<= 1) ss += __shfl_xor(ss, m, 32);
  const float rms = rsqrtf(ss * (1.0f / 128.0f) + 1e-5f);

  const float p = (float)pos[b];
  float tn[4] = {t.x, t.y, t.z, t.w};
#pragma unroll
  for (int s = 0; s < 4; ++s) tn[s] = tn[s] * rms * w[lane * 4 + s];

  float out[4];
#pragma unroll
  for (int s = 0; s < 4; ++s) {
    const int d = lane * 4 + s;
    const int j = d & 63;
    const float th = p * __powf(10000.0f, -(float)j * (1.0f / 64.0f));
    float sn, cs;
    __sincosf(th, &sn, &cs);
    const float partner = __shfl_xor(tn[s], 16, 32);  // element at d +/- 64
    out[s] = (d < 64) ? (tn[s] * cs - partner * sn)
                      : (tn[s] * cs + partner * sn);
  }
  *(float4*)(vec + lane * 4) = make_float4(out[0], out[1], out[2], out[3]);
}

// ---------------------------------------------------------------------------
// Flash-decode attention: one wave per (split, kv-head, batch).
// Q group (4 rows) padded to M=16 with zeros. 16-key tiles staged in LDS via
// CDNA5 async global->LDS b128 copies (no VGPR round-trip for the KV stream).
// ---------------------------------------------------------------------------
#define LDK 132   // padded row stride (floats): 16B-aligned rows, bank spread
#define LDP 20

__global__ void __launch_bounds__(32)
attn_split_kernel(const float* __restrict__ qb, const float* __restrict__ kb,
                  const float* __restrict__ vb,
                  const float* __restrict__ kcache,
                  const float* __restrict__ vcache,
                  float* __restrict__ Opart, float* __restrict__ Mpart,
                  float* __restrict__ Lpart) {
  __shared__ float ldsK[16 * LDK];
  __shared__ float ldsV[16 * LDK];
  __shared__ float ldsP[16 * LDP];

  const int lane  = threadIdx.x;
  const int l16   = lane & 15;
  const int half  = lane >> 4;
  const int split = blockIdx.x;
  const int h     = blockIdx.y;
  const int b     = blockIdx.z;
  const int bh    = b * HK_ + h;

  // Preload Q as A-matrix fragments for all 32 K-steps (K dim = 128).
  const float* qbase = qb + ((size_t)b * HQ_ + h * GQ_) * D_;
  v2f aq[32];
#pragma unroll
  for (int j = 0; j < 32; ++j) {
    const int kd = 4 * j + 2 * half;
    if (l16 < GQ_) {
      aq[j].x = qbase[l16 * D_ + kd];
      aq[j].y = qbase[l16 * D_ + kd + 1];
    } else {
      aq[j].x = 0.0f;
      aq[j].y = 0.0f;
    }
  }

  const float* kc = kcache + (size_t)bh * SPAST_ * D_;
  const float* vc = vcache + (size_t)bh * SPAST_ * D_;
  const float* kn = kb + (size_t)bh * D_;   // appended key  (index SPAST_)
  const float* vn = vb + (size_t)bh * D_;   // appended value

  const v8f zero8 = {0.f, 0.f, 0.f, 0.f, 0.f, 0.f, 0.f, 0.f};
  v8f O[8];
#pragma unroll
  for (int c = 0; c < 8; ++c) O[c] = zero8;
  float rowmax[8], rowsum[8];
#pragma unroll
  for (int r = 0; r < 8; ++r) { rowmax[r] = -INFINITY; rowsum[r] = 0.0f; }

  for (int t = 0; t < TILES_PER_SPLIT_; ++t) {
    const int s0 = (split * TILES_PER_SPLIT_ + t) * 16;

    // --- stage K/V tile (16 keys x 128 dims) into LDS: async b128 per lane ---
#pragma unroll
    for (int i = 0; i < 16; ++i) {
      const int f   = i * 32 + lane;  // 0..511 float4 slots
      const int row = f >> 5;
      const int c4  = f & 31;
      const int kg  = s0 + row;
      const float* ksrc =
          (kg < SPAST_) ? (kc + (size_t)kg * D_ + c4 * 4) : (kn + c4 * 4);
      const float* vsrc =
          (kg < SPAST_) ? (vc + (size_t)kg * D_ + c4 * 4) : (vn + c4 * 4);
      async_load_f4_to_lds((uint32_t)(uintptr_t)&ldsK[row * LDK + c4 * 4], ksrc);
      async_load_f4_to_lds((uint32_t)(uintptr_t)&ldsV[row * LDK + c4 * 4], vsrc);
    }
    wait_asynccnt0();  // data landed in LDS

    // --- scores: S(16x16) = Q(16x128) * K^T(128x16), 32 chained WMMAs ---
    v8f sc = zero8;
#pragma unroll
    for (int j = 0; j < 32; ++j) {
      const int kd = 4 * j + 2 * half;
      const float2 kk = *(const float2*)&ldsK[l16 * LDK + kd];
      v2f bb; bb.x = kk.x; bb.y = kk.y;
      sc = wmma_f32(aq[j], bb, sc);
    }

    // --- online softmax (per-row reductions across 16-lane halves) ---
    float fscale[8];
#pragma unroll
    for (int r = 0; r < 8; ++r) {
      float m = sc[r];
      m = fmaxf(m, __shfl_xor(m, 1, 32));
      m = fmaxf(m, __shfl_xor(m, 2, 32));
      m = fmaxf(m, __shfl_xor(m, 4, 32));
      m = fmaxf(m, __shfl_xor(m, 8, 32));
      const float newm = fmaxf(rowmax[r], m);
      const float f    = __expf(rowmax[r] - newm);
      const float pv   = __expf(sc[r] - newm);
      float ps = pv;
      ps += __shfl_xor(ps, 1, 32);
      ps += __shfl_xor(ps, 2, 32);
      ps += __shfl_xor(ps, 4, 32);
      ps += __shfl_xor(ps, 8, 32);
      rowsum[r] = rowsum[r] * f + ps;
      rowmax[r] = newm;
      fscale[r] = f;
      sc[r]     = pv;  // reuse as probability
    }
#pragma unroll
    for (int c = 0; c < 8; ++c)
#pragma unroll
      for (int r = 0; r < 8; ++r) O[c][r] *= fscale[r];

    // --- relayout P (C-frag -> A-frag) via LDS ---
#pragma unroll
    for (int r = 0; r < 8; ++r) ldsP[(r + 8 * half) * LDP + l16] = sc[r];
    __syncthreads();

    // --- O += P(16x16) * V(16x128): 8 N-chunks x 4 K-steps = 32 WMMAs ---
#pragma unroll
    for (int j = 0; j < 4; ++j) {
      const int kk = 4 * j + 2 * half;
      const float2 pa = *(const float2*)&ldsP[l16 * LDP + kk];
      v2f ap; ap.x = pa.x; ap.y = pa.y;
#pragma unroll
      for (int c = 0; c < 8; ++c) {
        v2f bv;
        bv.x = ldsV[kk * LDK + c * 16 + l16];
        bv.y = ldsV[(kk + 1) * LDK + c * 16 + l16];
        O[c] = wmma_f32(ap, bv, O[c]);
      }
    }
    // drain DS reads before async engine overwrites LDS next iteration
    // (async LDS writes are unordered vs ds ops, ISA 08 §4.1)
    __syncthreads();
  }

  // --- write split partials (only rows 0..3 are real; they live in half 0) ---
  if (half == 0) {
    const size_t obase = ((size_t)bh * NSPLIT_ + split) * (GQ_ * D_);
#pragma unroll
    for (int c = 0; c < 8; ++c)
#pragma unroll
      for (int r = 0; r < GQ_; ++r)
        Opart[obase + r * D_ + c * 16 + l16] = O[c][r];
  }
  if (lane == 0) {
    const size_t sb = ((size_t)bh * NSPLIT_ + split) * GQ_;
#pragma unroll
    for (int r = 0; r < GQ_; ++r) {
      Mpart[sb + r] = rowmax[r];
      Lpart[sb + r] = rowsum[r];
    }
  }
}

// ---------------------------------------------------------------------------
// Log-sum-exp combine across splits. grid = (HK, B), block = 128 (one dim/thr)
// ---------------------------------------------------------------------------
__global__ void __launch_bounds__(128)
attn_combine_kernel(const float* __restrict__ Opart,
                    const float* __restrict__ Mpart,
                    const float* __restrict__ Lpart,
                    float* __restrict__ attnbuf) {
  const int h  = blockIdx.x;
  const int b  = blockIdx.y;
  const int d  = threadIdx.x;
  const int bh = b * HK_ + h;
#pragma unroll
  for (int r = 0; r < GQ_; ++r) {
    float mstar = -INFINITY;
    for (int s = 0; s < NSPLIT_; ++s)
      mstar = fmaxf(mstar, Mpart[((size_t)bh * NSPLIT_ + s) * GQ_ + r]);
    float denom = 0.0f, acc = 0.0f;
    for (int s = 0; s < NSPLIT_; ++s) {
      const size_t sb = ((size_t)bh * NSPLIT_ + s);
      const float e = __expf(Mpart[sb * GQ_ + r] - mstar);
      denom += Lpart[sb * GQ_ + r] * e;
      acc   += Opart[sb * (GQ_ * D_) + r * D_ + d] * e;
    }
    attnbuf[((size_t)b * HQ_ + h * GQ_ + r) * D_ + d] = acc / denom;
  }
}

// ---------------------------------------------------------------------------
// Host launcher
// ---------------------------------------------------------------------------
extern "C" void kernel_launch(void* const* d_in, const int* in_sizes, int n_in,
                              void* d_out, int out_size, void* d_ws,
                              size_t ws_size, hipStream_t stream) {
  const float* x      = (const float*)d_in[0];
  const int*   pos    = (const int*)d_in[1];   // jax demotes int64 -> int32
  const float* kcache = (const float*)d_in[2];
  const float* vcache = (const float*)d_in[3];
  const float* Wq     = (const float*)d_in[4];
  const float* Wk     = (const float*)d_in[5];
  const float* Wv     = (const float*)d_in[6];
  const float* Wo     = (const float*)d_in[7];
  const float* qnw    = (const float*)d_in[8];
  const float* knw    = (const float*)d_in[9];
  float* out = (float*)d_out;

  float* ws = (float*)d_ws;
  size_t off = 0;
  float* qbuf    = ws + off; off += (size_t)B_ * HQ_ * D_;
  float* kbuf    = ws + off; off += (size_t)B_ * HK_ * D_;
  float* vbuf    = ws + off; off += (size_t)B_ * HK_ * D_;
  float* attnbuf = ws + off; off += (size_t)B_ * HQ_ * D_;
  float* Opart   = ws + off; off += (size_t)B_ * HK_ * NSPLIT_ * GQ_ * D_;
  float* Mpart   = ws + off; off += (size_t)B_ * HK_ * NSPLIT_ * GQ_;
  float* Lpart   = ws + off; off += (size_t)B_ * HK_ * NSPLIT_ * GQ_;

  // 1) QKV projections (fp32 WMMA GEMM, weights are L2-resident)
  gemm16_tn<<<dim3(DIM_ / 16, B_ / 16), 32, 0, stream>>>(x, Wq, qbuf, B_, DIM_, DIM_);
  gemm16_tn<<<dim3((HK_ * D_) / 16, B_ / 16), 32, 0, stream>>>(x, Wk, kbuf, B_, HK_ * D_, DIM_);
  gemm16_tn<<<dim3((HK_ * D_) / 16, B_ / 16), 32, 0, stream>>>(x, Wv, vbuf, B_, HK_ * D_, DIM_);

  // 2) RMSNorm + RoPE (in place on q/k projections)
  normrope_kernel<<<dim3(B_, HQ_ + HK_), 32, 0, stream>>>(qbuf, kbuf, qnw, knw, pos);

  // 3) Flash-decode over 8 key splits (2048 waves streaming the KV cache)
  attn_split_kernel<<<dim3(NSPLIT_, HK_, B_), 32, 0, stream>>>(
      qbuf, kbuf, vbuf, kcache, vcache, Opart, Mpart, Lpart);

  // 4) LSE combine
  attn_combine_kernel<<<dim3(HK_, B_), 128, 0, stream>>>(Opart, Mpart, Lpart, attnbuf);

  // 5) Output projection
  gemm16_tn<<<dim3(DIM_ / 16, B_ / 16), 32, 0, stream>>>(attnbuf, Wo, out, B_, DIM_, DIM_);
}